// CustomQueryTransform_89919435309261
// MI455X (gfx1250) — compile-verified
//
#include <hip/hip_runtime.h>

typedef __attribute__((ext_vector_type(2))) float v2f;
typedef __attribute__((ext_vector_type(8))) float v8f;
typedef __attribute__((ext_vector_type(4))) unsigned int u32x4;
typedef __attribute__((ext_vector_type(4))) int i32x4;
typedef __attribute__((ext_vector_type(8))) int i32x8;

#define DD 1024          // feature dim
#define TT 2048          // time steps
#define BB 8             // batch
#define NROW (BB * TT)   // 16384 rows of X/U

// LDS A-tile geometry: 16 rows x 512 dwords, TDM-padded 2 dwords per 256 dwords
// -> row stride 516 dwords (== 4 mod 64 banks: conflict-free, 8B aligned)
#define RSTRIDE 516
#define CHUNK2  258

// ---------------------------------------------------------------------------
// Kernel A: U[n][o] = sum_d X[n][d] * Wcur[o][d]   (U written into d_out)
// One block per 16-row M tile. Wave 0 DMAs the X tile into LDS with the
// Tensor Data Mover (two 16x512 passes, TENSORcnt), all 8 waves consume it;
// each wave computes a 16(M) x 128(O) strip = 8 WMMA tiles.
// ---------------------------------------------------------------------------
__global__ __launch_bounds__(256)
void gemm_x_wcurT(const float* __restrict__ X, const float* __restrict__ W,
                  float* __restrict__ U) {
  __shared__ float atile[16 * RSTRIDE];   // ~32 KB

  const int tid  = threadIdx.x;
  const int lane = tid & 31;
  const int wave = tid >> 5;
  const int half = lane >> 4;
  const int l    = lane & 15;
  const int m0   = blockIdx.x * 16;

  const int o0w = wave * 128;             // this wave's 128-wide O strip
  const float* wrow = W + (size_t)(o0w + l) * DD + 2 * half;

  v8f acc[8];
#pragma unroll
  for (int j = 0; j < 8; ++j) acc[j] = (v8f){};

  for (int p = 0; p < 2; ++p) {           // two K halves of 512
    if (tid < 32) {
      // ---- build Tensor DMA descriptor (D#) and issue TDM load ----
      unsigned lds_off = (unsigned)(size_t)&atile[0];   // low 32 bits = LDS offset
      unsigned long long ga =
          (unsigned long long)(const void*)(X + (size_t)m0 * DD + p * 512);
      u32x4 g0 = {
        1u,                                              // count=1 user descriptor
        lds_off,                                         // lds_addr
        (unsigned)ga,                                    // global_addr[31:0]
        (unsigned)((ga >> 32) & 0x01FFFFFFu) | (2u << 30) // addr[56:32] | type=2
      };
      i32x8 g1 = {
        (int)0x03D20000,   // data_size=4B | pad_enable | interval=256dw | amount=2dw
        (int)0x04000000,   // tensor_dim0[15:0]=1024 in [63:48]
        (int)0x40000000,   // tensor_dim1[15:0]=16384 in [95:80]
        (int)0x02000000,   // tile_dim0=512 in [127:112]
        16,                // tile_dim1=16, tile_dim2=0
        1024,              // tensor_dim0_stride[31:0]
        0, 0
      };
      i32x4 gz4 = {0, 0, 0, 0};
#if __clang_major__ >= 23
      i32x8 gz8 = {0, 0, 0, 0, 0, 0, 0, 0};
      __builtin_amdgcn_tensor_load_to_lds(g0, g1, gz4, gz4, gz8, 0);
#else
      __builtin_amdgcn_tensor_load_to_lds(g0, g1, gz4, gz4, 0);
#endif
      __builtin_amdgcn_s_wait_tensorcnt(0);
    }
    __syncthreads();                      // LDS tile visible to all waves

#pragma unroll
    for (int c = 0; c < 2; ++c) {         // two 256-dword chunks in this half
      const float* ab = &atile[l * RSTRIDE + c * CHUNK2 + 2 * half];
      const float* wb = wrow + p * 512 + c * 256;
      for (int k0 = 0; k0 < 256; k0 += 4) {
        v2f a = *(const v2f*)(ab + k0);   // ds_load_b64, conflict-free
#pragma unroll
        for (int j = 0; j < 8; ++j) {
          v2f b = *(const v2f*)(wb + (size_t)j * 16 * DD + k0);
          acc[j] = __builtin_amdgcn_wmma_f32_16x16x4_f32(
              false, a, false, b, (short)0, acc[j], false, false);
        }
      }
    }
    __syncthreads();                      // done reading before next TDM overwrite
  }

#pragma unroll
  for (int j = 0; j < 8; ++j) {
#pragma unroll
    for (int i = 0; i < 8; ++i) {
      // C/D layout: VGPR i -> row i + 8*half, col l
      U[(size_t)(m0 + i + 8 * half) * DD + o0w + j * 16 + l] = acc[j][i];
    }
  }
}

// ---------------------------------------------------------------------------
// State init (carry c padded to 16 rows, barrier counter) — run every call so
// kernel_launch stays deterministic under graph replay.
// ---------------------------------------------------------------------------
__global__ void init_state(float* __restrict__ cbuf, unsigned* __restrict__ counter) {
  int i = blockIdx.x * blockDim.x + threadIdx.x;
  if (i < 16 * DD) cbuf[i] = 0.0f;
  if (i == 0) *counter = 0u;
}

// ---------------------------------------------------------------------------
// Kernel B: persistent sequential scan.
//   y_t = u_t + c @ Wcum^T ;  c += y_t
// 64 blocks, block k owns output features [16k, 16k+16). Per step each block
// computes a 16x16 tile: K=1024 split over 8 waves (32 WMMAs each), partials
// reduced via ds_add_f32 into LDS, then y overwrites u in d_out and the c
// slice is updated. Device-wide generation barrier between steps.
// ---------------------------------------------------------------------------
__global__ __launch_bounds__(256)
void scan_persistent(const float* __restrict__ Wc,   // W_cum (1024x1024)
                     float* __restrict__ out,        // holds U, overwritten with Y
                     float* __restrict__ cbuf,       // 16 x 1024 carry
                     unsigned* __restrict__ counter) {
  __shared__ float tile[256];

  const int tid  = threadIdx.x;
  const int lane = tid & 31;
  const int wave = tid >> 5;
  const int half = lane >> 4;
  const int l    = lane & 15;
  const int o0   = blockIdx.x * 16;
  const int kb   = wave * 128;                 // this wave's K slice

  const float* wrow = Wc   + (size_t)(o0 + l) * DD + 2 * half + kb;
  const float* crow = cbuf + (size_t)l * DD + 2 * half + kb;

  const int m = tid >> 4;   // 0..15 : padded batch row for elementwise work
  const int n = tid & 15;   // 0..15 : feature within this block's slice

  const unsigned nblk = gridDim.x;

  for (int t = 0; t < TT; ++t) {
    // ---- partial GEMM: this wave's K slice of c(16x1024) x Wcum^T tile ----
    v8f acc = (v8f){};
#pragma unroll 8
    for (int k0 = 0; k0 < 128; k0 += 4) {
      v2f a = *(const v2f*)(crow + k0);
      v2f b = *(const v2f*)(wrow + k0);
      acc = __builtin_amdgcn_wmma_f32_16x16x4_f32(
          false, a, false, b, (short)0, acc, false, false);
    }

    // ---- seed LDS tile with u_t (pad rows 8..15 are zero) ----
    float u = 0.0f;
    if (m < BB) u = out[(size_t)(m * TT + t) * DD + o0 + n];
    tile[tid] = u;
    __syncthreads();

    // ---- reduce 8 wave partials with LDS float atomics (ds_add_f32) ----
#pragma unroll
    for (int i = 0; i < 8; ++i)
      atomicAdd(&tile[(i + 8 * half) * 16 + l], acc[i]);
    __syncthreads();

    // ---- y, output store, carry update (each thread owns one element) ----
    float y  = tile[m * 16 + n];
    float cn = cbuf[(size_t)m * DD + o0 + n] + y;
    if (m < BB) out[(size_t)(m * TT + t) * DD + o0 + n] = y;
    cbuf[(size_t)m * DD + o0 + n] = cn;

    // ---- device-wide generation barrier ----
    __threadfence();        // publish c slice
    __syncthreads();
    if (tid == 0) {
      atomicAdd(counter, 1u);
      const unsigned target = (unsigned)(t + 1) * nblk;
      while (__hip_atomic_load(counter, __ATOMIC_RELAXED, __HIP_MEMORY_SCOPE_AGENT) < target)
        __builtin_amdgcn_s_sleep(2);
    }
    __syncthreads();
    __threadfence();        // acquire other blocks' c before next step
  }
}

// ---------------------------------------------------------------------------
extern "C" void kernel_launch(void* const* d_in, const int* in_sizes, int n_in,
                              void* d_out, int out_size, void* d_ws, size_t ws_size,
                              hipStream_t stream) {
  const float* queries = (const float*)d_in[0];   // (8,2048,1,1024) fp32
  const float* W_cur   = (const float*)d_in[1];   // (1024,1024)
  const float* W_cum   = (const float*)d_in[2];   // (1024,1024)
  float* out = (float*)d_out;                     // (8,2048,1,1024)

  float*    cbuf    = (float*)d_ws;                                   // 64 KB
  unsigned* counter = (unsigned*)((char*)d_ws + 16 * DD * sizeof(float));

  init_state<<<64, 256, 0, stream>>>(cbuf, counter);
  gemm_x_wcurT<<<NROW / 16, 256, 0, stream>>>(queries, W_cur, out);   // 1024 blocks
  scan_persistent<<<DD / 16, 256, 0, stream>>>(W_cum, out, cbuf, counter); // 64 blocks
}